// Q_V_network_44616120271394
// MI455X (gfx1250) — compile-verified
//
#include <hip/hip_runtime.h>
#include <hip/hip_bf16.h>

// ---------------------------------------------------------------------------
// CDNA5 / gfx1250 implementation of the multi-agent Q/V network.
// All GEMMs run on V_WMMA_F32_16X16X4_F32 (fp32-exact vs the reference).
// One wave == one 16x16 C tile; 4 waves per workgroup share one B column
// tile, staged once into LDS in packed fragment order (K-pairs contiguous)
// so every B fragment is a single aligned ds_load_b64.
// ---------------------------------------------------------------------------

typedef float v2f __attribute__((ext_vector_type(2)));
typedef float v8f __attribute__((ext_vector_type(8)));

#if defined(__HIP_DEVICE_COMPILE__) && !__has_builtin(__builtin_amdgcn_wmma_f32_16x16x4_f32)
#error "wmma f32 16x16x4 builtin not available for this offload target"
#endif

__device__ __forceinline__ float gelu_exact(float x) {
    return 0.5f * x * (1.0f + erff(x * 0.70710678118654752440f));
}

__device__ __forceinline__ float sigmoidf(float x) {
    return 1.0f / (1.0f + expf(-x));
}

// Stage a K x 16 B tile (column block col0) into LDS, packed so that the
// (K, K+1) pair for column n is contiguous: bp[k/2][n][0..1].
template <int K>
__device__ __forceinline__ void stage_bpack(const float* __restrict__ B, int ldb,
                                            int col0, float (*bp)[16][2]) {
#pragma unroll
    for (int p = threadIdx.x; p < (K / 2) * 16; p += 128) {
        const int kp = p >> 4;
        const int n = p & 15;
        bp[kp][n][0] = B[(size_t)(2 * kp) * ldb + col0 + n];
        bp[kp][n][1] = B[(size_t)(2 * kp + 1) * ldb + col0 + n];
    }
}

// One wave: 16x16 fp32 tile of A(16xK) @ B(KxN tile in packed LDS), K%4==0.
// Fragment layout per ISA 7.12.2 (32-bit A 16x4): lanes 0-15 hold M=lane with
// K+0/K+1 in the two A VGPRs; lanes 16-31 hold K+2/K+3.  B symmetric, lane=N.
template <int K>
__device__ __forceinline__ v8f wmma_tile16_lds(const float* __restrict__ A, int lda,
                                               const float (*bp)[16][2], int row0) {
    const int lane = threadIdx.x & 31;
    const int m = lane & 15;
    const int half = lane >> 4;
    const int koff = half << 1;          // 0 or 2
    v8f acc = {0.f, 0.f, 0.f, 0.f, 0.f, 0.f, 0.f, 0.f};
    const float* Ar = A + (size_t)(row0 + m) * lda + koff;
#pragma unroll
    for (int k0 = 0; k0 < K; k0 += 4) {
        const v2f a = *(const v2f*)(Ar + k0);                  // global_load_b64
        const v2f b = *(const v2f*)(bp[(k0 >> 1) + half][m]);  // ds_load_b64
        acc = __builtin_amdgcn_wmma_f32_16x16x4_f32(false, a, false, b,
                                                    (short)0, acc, false, false);
    }
    return acc;
}

// ---------------------------------------------------------------------------
// Generic row-parallel GEMM: C[z] = epilogue(A @ B[z] + bias[z])
//   A = [A0 | A1] concatenated at column KSPLIT (KSPLIT == K -> A0 only).
//   grid = (rowTiles/4, colTiles, Z); block = 128 (4 waves, one tile each;
//   all four waves share the same B tile staged in LDS).
//   EP: 0 = none, 2 = bias + gelu.
// ---------------------------------------------------------------------------
template <int K, int KSPLIT, int EP>
__global__ __launch_bounds__(128) void gemm_f32_kernel(
        const float* __restrict__ A0, int lda0,
        const float* __restrict__ A1, int lda1, int zStrideA,
        const float* __restrict__ B, int ldb, int zStrideB,
        const float* __restrict__ bias, int zStrideBias,
        float* __restrict__ C, int ldc, int zStrideC) {
    const int wave = threadIdx.x >> 5;
    const int lane = threadIdx.x & 31;
    const int m = lane & 15;
    const int half = lane >> 4;
    const int koff = half << 1;          // 0 or 2
    const int row0 = (blockIdx.x * 4 + wave) * 16;
    const int col0 = blockIdx.y * 16;
    const int z = blockIdx.z;

    const float* A0z = A0 + (size_t)z * zStrideA;
    const float* Bz = B + (size_t)z * zStrideB;

    __shared__ float bp[K / 2][16][2];
    __builtin_prefetch(A0z + (size_t)row0 * lda0, 0, 1);   // global_prefetch_b8
    stage_bpack<K>(Bz, ldb, col0, bp);
    __syncthreads();

    v8f acc = {0.f, 0.f, 0.f, 0.f, 0.f, 0.f, 0.f, 0.f};

    {   // A0 part: columns [0, KSPLIT)
        const float* Ar = A0z + (size_t)(row0 + m) * lda0 + koff;
#pragma unroll
        for (int k0 = 0; k0 < KSPLIT; k0 += 4) {
            const v2f a = *(const v2f*)(Ar + k0);
            const v2f b = *(const v2f*)(bp[(k0 >> 1) + half][m]);
            acc = __builtin_amdgcn_wmma_f32_16x16x4_f32(false, a, false, b,
                                                        (short)0, acc, false, false);
        }
    }
    if constexpr (KSPLIT < K) {   // A1 part: columns [KSPLIT, K)
        const float* Ar = A1 + (size_t)(row0 + m) * lda1 + koff;
#pragma unroll
        for (int k0 = KSPLIT; k0 < K; k0 += 4) {
            const v2f a = *(const v2f*)(Ar + (k0 - KSPLIT));
            const v2f b = *(const v2f*)(bp[(k0 >> 1) + half][m]);
            acc = __builtin_amdgcn_wmma_f32_16x16x4_f32(false, a, false, b,
                                                        (short)0, acc, false, false);
        }
    }

    const int n = col0 + m;
    float bv = 0.0f;
    if constexpr (EP != 0) bv = bias[(size_t)z * zStrideBias + n];
    float* Cz = C + (size_t)z * zStrideC;
#pragma unroll
    for (int v = 0; v < 8; ++v) {
        const int r = row0 + v + (half << 3);
        float x = acc[v] + bv;
        if constexpr (EP == 2) x = gelu_exact(x);
        Cz[(size_t)r * ldc + n] = x;
    }
}

// ---------------------------------------------------------------------------
// OAsum[b][d] = sum_j OA[b*32+j][d]
// ---------------------------------------------------------------------------
__global__ void oasum_kernel(const float* __restrict__ OA, float* __restrict__ OAs) {
    const int b = blockIdx.x;
    const int d = threadIdx.x;   // 64 threads
    float s = 0.0f;
    for (int j = 0; j < 32; ++j) s += OA[((size_t)(b * 32 + j)) * 64 + d];
    OAs[b * 64 + d] = s;
}

// ---------------------------------------------------------------------------
// Attention / gate / softmax / aggregation / LayerNorms.
// 4 waves per block; each wave handles one (b, a) pair (lane k = neighbor k).
// Writes weights_full (output) and aggf into cat[:, 64:128].
// ---------------------------------------------------------------------------
__global__ __launch_bounds__(128) void attn_kernel(
        const float* __restrict__ Qh, const float* __restrict__ Kh,
        const float* __restrict__ AV,
        const float* __restrict__ qpart, const float* __restrict__ kpart,
        const float* __restrict__ bha,
        const float* __restrict__ W_hal, const float* __restrict__ b_hal,
        const float* __restrict__ gum,
        const float* __restrict__ OA, const float* __restrict__ OAs,
        const float* __restrict__ ln1_g, const float* __restrict__ ln1_b,
        const float* __restrict__ W_avl, const float* __restrict__ b_avl,
        const float* __restrict__ ln2_g, const float* __restrict__ ln2_b,
        float* __restrict__ wf, float* __restrict__ cat) {
    const int wave = threadIdx.x >> 5;
    const int lane = threadIdx.x & 31;
    const int g = blockIdx.x * 4 + wave;     // 0..4095
    const int b = g >> 5;
    const int a = g & 31;
    const int row = g;

    __shared__ float s_sc[4][4][32];
    __shared__ float s_w[4][4][32];
    __shared__ float s_mx[4][4], s_sum[4][4];
    __shared__ float s_x[4][64];
    __shared__ float s_stat[4][2];

    if (lane < 31) {
        const int k = lane;
        const int j = k + (k >= a);
        const int rj = (b << 5) + j;
        // inter -> logits for the Gumbel hard gate
        float l0 = b_hal[0], l1 = b_hal[1];
        for (int h = 0; h < 4; ++h) {
            const float* qp = qpart + ((size_t)h * 4096 + row) * 16;
            const float* kp = kpart + ((size_t)h * 4096 + rj) * 16;
            const float* bh = bha + h * 16;
#pragma unroll
            for (int f = 0; f < 16; ++f) {
                const float t = gelu_exact(qp[f] + kp[f] + bh[f]);
                const int d = h * 16 + f;
                l0 += t * W_hal[d * 2 + 0];
                l1 += t * W_hal[d * 2 + 1];
            }
        }
        const float* gn = gum + ((size_t)row * 31 + k) * 2;
        // argmax of softmax((logits+g)/tau) == argmax(logits+g); ties -> idx 0
        const float hw = ((l1 + gn[1]) > (l0 + gn[0])) ? 1.0f : 0.0f;
        for (int h = 0; h < 4; ++h) {
            float sc = 0.0f;
            if (hw > 0.0f) {
                const float* q = Qh + ((size_t)h * 4096 + row) * 64;
                const float* kv = Kh + ((size_t)h * 4096 + rj) * 64;
                float d = 0.0f;
                for (int e = 0; e < 64; ++e) d += q[e] * kv[e];
                sc = d * 0.125f;    // 1/sqrt(64)
            }
            s_sc[wave][h][k] = sc;
        }
    }
    __syncthreads();

    if (lane < 4) {
        float mx = -1e30f;
        for (int k = 0; k < 31; ++k) mx = fmaxf(mx, s_sc[wave][lane][k]);
        float sm = 0.0f;
        for (int k = 0; k < 31; ++k) sm += expf(s_sc[wave][lane][k] - mx);
        s_mx[wave][lane] = mx;
        s_sum[wave][lane] = sm;
    }
    __syncthreads();

    if (lane < 31) {
        const int k = lane;
        const int j = k + (k >= a);
        for (int h = 0; h < 4; ++h) {
            const float w = expf(s_sc[wave][h][k] - s_mx[wave][h]) / s_sum[wave][h];
            s_w[wave][h][k] = w;
            wf[(((size_t)b * 4 + h) * 32 + a) * 32 + j] = w;
        }
    } else {   // lane 31: diagonal = 1
        for (int h = 0; h < 4; ++h)
            wf[(((size_t)b * 4 + h) * 32 + a) * 32 + a] = 1.0f;
    }
    __syncthreads();

    // agg[d] + (sum_oa)  -> LN1 input
    for (int d = lane; d < 64; d += 32) {
        const int h = d >> 4, f = d & 15;
        float acc = 0.0f;
        for (int k = 0; k < 31; ++k) {
            const int j = k + (k >= a);
            const int rj = (b << 5) + j;
            acc += s_w[wave][h][k] * AV[((size_t)h * 4096 + rj) * 16 + f];
        }
        s_x[wave][d] = OAs[b * 64 + d] - OA[(size_t)row * 64 + d] + acc;
    }
    __syncthreads();

    if (lane == 0) {
        float m = 0.0f;
        for (int d = 0; d < 64; ++d) m += s_x[wave][d];
        m *= (1.0f / 64.0f);
        float v = 0.0f;
        for (int d = 0; d < 64; ++d) { const float t = s_x[wave][d] - m; v += t * t; }
        v *= (1.0f / 64.0f);
        s_stat[wave][0] = m;
        s_stat[wave][1] = rsqrtf(v + 1e-5f);
    }
    __syncthreads();

    for (int d = lane; d < 64; d += 32)
        s_x[wave][d] = (s_x[wave][d] - s_stat[wave][0]) * s_stat[wave][1] * ln1_g[d] + ln1_b[d];
    __syncthreads();

    // agg_ + gelu(agg_ @ W_avl + b_avl)
    float yv[2];
#pragma unroll
    for (int i = 0; i < 2; ++i) {
        const int d = lane + 32 * i;
        float acc = b_avl[d];
        for (int k2 = 0; k2 < 64; ++k2) acc += s_x[wave][k2] * W_avl[k2 * 64 + d];
        yv[i] = s_x[wave][d] + gelu_exact(acc);
    }
    __syncthreads();
    s_x[wave][lane] = yv[0];
    s_x[wave][lane + 32] = yv[1];
    __syncthreads();

    if (lane == 0) {
        float m = 0.0f;
        for (int d = 0; d < 64; ++d) m += s_x[wave][d];
        m *= (1.0f / 64.0f);
        float v = 0.0f;
        for (int d = 0; d < 64; ++d) { const float t = s_x[wave][d] - m; v += t * t; }
        v *= (1.0f / 64.0f);
        s_stat[wave][0] = m;
        s_stat[wave][1] = rsqrtf(v + 1e-5f);
    }
    __syncthreads();

    for (int d = lane; d < 64; d += 32) {
        const float o = (s_x[wave][d] - s_stat[wave][0]) * s_stat[wave][1] * ln2_g[d] + ln2_b[d];
        cat[(size_t)row * 128 + 64 + d] = o;   // aggf half of [sq | aggf]
    }
}

// ---------------------------------------------------------------------------
// Fully fused GRU: 6 WMMA tile GEMMs (xr,xz,xn,hr,hz,hn) + gate epilogue.
// grid = (rowTiles/4, 4 col tiles); block = 128.  The 6 shared B tiles
// (W_ih/W_hh at cols c0, 64+c0, 128+c0) are staged packed in LDS (24 KB).
// ---------------------------------------------------------------------------
__global__ __launch_bounds__(128) void gru_kernel(
        const float* __restrict__ feat, const float* __restrict__ hist,
        const float* __restrict__ W_ih, const float* __restrict__ b_ih,
        const float* __restrict__ W_hh, const float* __restrict__ b_hh,
        float* __restrict__ hnew) {
    const int wave = threadIdx.x >> 5;
    const int row0 = (blockIdx.x * 4 + wave) * 16;
    const int c0 = blockIdx.y * 16;

    __shared__ float bp[6][32][16][2];
#pragma unroll
    for (int p = threadIdx.x; p < 6 * 32 * 16; p += 128) {
        const int t = p >> 9;            // tile 0..5
        const int kp = (p >> 4) & 31;
        const int n = p & 15;
        const float* Bsrc = (t < 3) ? W_ih : W_hh;
        const int cc = c0 + (t % 3) * 64;
        bp[t][kp][n][0] = Bsrc[(size_t)(2 * kp) * 192 + cc + n];
        bp[t][kp][n][1] = Bsrc[(size_t)(2 * kp + 1) * 192 + cc + n];
    }
    __syncthreads();

    const v8f xr = wmma_tile16_lds<64>(feat, 64, bp[0], row0);
    const v8f xz = wmma_tile16_lds<64>(feat, 64, bp[1], row0);
    const v8f xn = wmma_tile16_lds<64>(feat, 64, bp[2], row0);
    const v8f hr = wmma_tile16_lds<64>(hist, 64, bp[3], row0);
    const v8f hz = wmma_tile16_lds<64>(hist, 64, bp[4], row0);
    const v8f hn = wmma_tile16_lds<64>(hist, 64, bp[5], row0);

    const int lane = threadIdx.x & 31;
    const int m = lane & 15;
    const int half = lane >> 4;
    const int c = c0 + m;
#pragma unroll
    for (int v = 0; v < 8; ++v) {
        const int r = row0 + v + (half << 3);
        const float rr = sigmoidf(xr[v] + b_ih[c] + hr[v] + b_hh[c]);
        const float zz = sigmoidf(xz[v] + b_ih[64 + c] + hz[v] + b_hh[64 + c]);
        const float nn = tanhf(xn[v] + b_ih[128 + c] + rr * (hn[v] + b_hh[128 + c]));
        const float hv = hist[(size_t)r * 64 + c];
        hnew[(size_t)r * 64 + c] = (1.0f - zz) * nn + zz * hv;
    }
}

// ---------------------------------------------------------------------------
// Output heads: Q = Qhid @ W_q2 + b_q2 (WMMA, N=16), Qv = sum(actions*Q);
// V = Vhid @ W_v2 + b_v2 (length-64 dot per row).  4 waves per block.
// ---------------------------------------------------------------------------
__global__ __launch_bounds__(128) void heads_kernel(
        const float* __restrict__ Qhid, const float* __restrict__ Vhid,
        const float* __restrict__ W_q2, const float* __restrict__ b_q2,
        const float* __restrict__ actions,
        const float* __restrict__ W_v2, const float* __restrict__ b_v2,
        float* __restrict__ Qv, float* __restrict__ V) {
    const int wave = threadIdx.x >> 5;
    const int lane = threadIdx.x & 31;
    const int row0 = (blockIdx.x * 4 + wave) * 16;
    const int m = lane & 15;
    const int half = lane >> 4;

    __shared__ float bq[32][16][2];
    __shared__ float Qt[4][16][16];
    __shared__ float vp[4][2][16];

    stage_bpack<64>(W_q2, 16, 0, bq);
    __syncthreads();

    const v8f q = wmma_tile16_lds<64>(Qhid, 64, bq, row0);
#pragma unroll
    for (int v = 0; v < 8; ++v)
        Qt[wave][v + (half << 3)][m] = q[v] + b_q2[m];

    // V partial dot: lane (half,m) covers K range [half*32, half*32+32)
    {
        const float* vr = Vhid + (size_t)(row0 + m) * 64;
        float acc = 0.0f;
        for (int k = half * 32; k < half * 32 + 32; ++k) acc += vr[k] * W_v2[k];
        vp[wave][half][m] = acc;
    }
    __syncthreads();

    if (lane < 16) {
        const int r = row0 + lane;
        float s = 0.0f;
#pragma unroll
        for (int c = 0; c < 16; ++c) s += actions[(size_t)r * 16 + c] * Qt[wave][lane][c];
        Qv[r] = s;
        V[r] = vp[wave][0][lane] + vp[wave][1][lane] + b_v2[0];
    }
}

// ---------------------------------------------------------------------------
// Host side
// ---------------------------------------------------------------------------
extern "C" void kernel_launch(void* const* d_in, const int* in_sizes, int n_in,
                              void* d_out, int out_size, void* d_ws, size_t ws_size,
                              hipStream_t stream) {
    const float* states  = (const float*)d_in[0];   // (128,32,128)
    const float* history = (const float*)d_in[1];   // (128,32,64)
    const float* actions = (const float*)d_in[2];   // (128,32,16)
    const float* gum     = (const float*)d_in[3];   // (128,32,31,2)
    const float* W_se  = (const float*)d_in[4];
    const float* b_se  = (const float*)d_in[5];
    const float* W_sae = (const float*)d_in[6];
    const float* b_sae = (const float*)d_in[7];
    const float* Wk    = (const float*)d_in[8];
    const float* bk    = (const float*)d_in[9];
    const float* Wq    = (const float*)d_in[10];
    const float* bq    = (const float*)d_in[11];
    const float* Wav   = (const float*)d_in[12];
    const float* bav   = (const float*)d_in[13];
    const float* Wha   = (const float*)d_in[14];
    const float* bha   = (const float*)d_in[15];
    const float* W_hal = (const float*)d_in[16];
    const float* b_hal = (const float*)d_in[17];
    const float* ln1_g = (const float*)d_in[18];
    const float* ln1_b = (const float*)d_in[19];
    const float* W_avl = (const float*)d_in[20];
    const float* b_avl = (const float*)d_in[21];
    const float* ln2_g = (const float*)d_in[22];
    const float* ln2_b = (const float*)d_in[23];
    const float* W_com = (const float*)d_in[24];
    const float* b_com = (const float*)d_in[25];
    const float* W_ih  = (const float*)d_in[26];
    const float* b_ih  = (const float*)d_in[27];
    const float* W_hh  = (const float*)d_in[28];
    const float* b_hh  = (const float*)d_in[29];
    const float* W_q1  = (const float*)d_in[30];
    const float* b_q1  = (const float*)d_in[31];
    const float* W_q2  = (const float*)d_in[32];
    const float* b_q2  = (const float*)d_in[33];
    const float* W_v1  = (const float*)d_in[34];
    const float* b_v1  = (const float*)d_in[35];
    const float* W_v2  = (const float*)d_in[36];
    const float* b_v2  = (const float*)d_in[37];

    const int R = 128 * 32;        // 4096 rows

    float* ws = (float*)d_ws;
    size_t o = 0;
    float* cat   = ws + o; o += (size_t)R * 128;        // [sq | aggf]
    float* OA    = ws + o; o += (size_t)R * 64;
    float* OAs   = ws + o; o += 128 * 64;
    float* Khb   = ws + o; o += (size_t)4 * R * 64;
    float* Qhb   = ws + o; o += (size_t)4 * R * 64;
    float* AVb   = ws + o; o += (size_t)4 * R * 16;
    float* qpb   = ws + o; o += (size_t)4 * R * 16;
    float* kpb   = ws + o; o += (size_t)4 * R * 16;
    float* featb = ws + o; o += (size_t)R * 64;
    float* Qhidb = ws + o; o += (size_t)R * 64;
    float* Vhidb = ws + o; o += (size_t)R * 64;

    float* out  = (float*)d_out;
    float* Qv   = out;                     // 4096
    float* V    = out + 4096;              // 4096
    float* hnew = out + 8192;              // 262144
    float* wf   = out + 8192 + (size_t)R * 64;  // weights_full 524288

    const dim3 blk(128);
    const int bt = (R / 16) / 4;   // 64 blocks of 4 row-tile waves

    // sq = gelu(states @ W_se + b_se) -> cat[:, 0:64]
    gemm_f32_kernel<128, 128, 2><<<dim3(bt, 4, 1), blk, 0, stream>>>(
        states, 128, states, 128, 0, W_se, 64, 0, b_se, 0, cat, 128, 0);
    // OA = gelu([states|actions] @ W_sae + b_sae)
    gemm_f32_kernel<144, 128, 2><<<dim3(bt, 4, 1), blk, 0, stream>>>(
        states, 128, actions, 16, 0, W_sae, 64, 0, b_sae, 0, OA, 64, 0);
    // per-head K = gelu(sq @ Wk[h] + bk[h])
    gemm_f32_kernel<64, 64, 2><<<dim3(bt, 4, 4), blk, 0, stream>>>(
        cat, 128, cat, 128, 0, Wk, 64, 64 * 64, bk, 64, Khb, 64, R * 64);
    // per-head Q = gelu(sq @ Wq[h] + bq[h])
    gemm_f32_kernel<64, 64, 2><<<dim3(bt, 4, 4), blk, 0, stream>>>(
        cat, 128, cat, 128, 0, Wq, 64, 64 * 64, bq, 64, Qhb, 64, R * 64);
    // per-head AV = gelu(OA @ Wav[h] + bav[h])
    gemm_f32_kernel<64, 64, 2><<<dim3(bt, 1, 4), blk, 0, stream>>>(
        OA, 64, OA, 64, 0, Wav, 16, 64 * 16, bav, 16, AVb, 16, R * 16);
    // qpart = Q[h] @ Wha[h][:64]  (no bias/activation here)
    gemm_f32_kernel<64, 64, 0><<<dim3(bt, 1, 4), blk, 0, stream>>>(
        Qhb, 64, Qhb, 64, R * 64, Wha, 16, 128 * 16, nullptr, 0, qpb, 16, R * 16);
    // kpart = K[h] @ Wha[h][64:]
    gemm_f32_kernel<64, 64, 0><<<dim3(bt, 1, 4), blk, 0, stream>>>(
        Khb, 64, Khb, 64, R * 64, Wha + 64 * 16, 16, 128 * 16, nullptr, 0, kpb, 16, R * 16);

    oasum_kernel<<<128, 64, 0, stream>>>(OA, OAs);

    attn_kernel<<<R / 4, blk, 0, stream>>>(Qhb, Khb, AVb, qpb, kpb, bha, W_hal, b_hal, gum,
                                           OA, OAs, ln1_g, ln1_b, W_avl, b_avl, ln2_g, ln2_b,
                                           wf, cat);

    // feat = gelu([sq | aggf] @ W_com + b_com)
    gemm_f32_kernel<128, 128, 2><<<dim3(bt, 4, 1), blk, 0, stream>>>(
        cat, 128, cat, 128, 0, W_com, 64, 0, b_com, 0, featb, 64, 0);

    gru_kernel<<<dim3(bt, 4), blk, 0, stream>>>(featb, history, W_ih, b_ih, W_hh, b_hh, hnew);

    // Q/V hidden layers
    gemm_f32_kernel<64, 64, 2><<<dim3(bt, 4, 1), blk, 0, stream>>>(
        hnew, 64, hnew, 64, 0, W_q1, 64, 0, b_q1, 0, Qhidb, 64, 0);
    gemm_f32_kernel<64, 64, 2><<<dim3(bt, 4, 1), blk, 0, stream>>>(
        hnew, 64, hnew, 64, 0, W_v1, 64, 0, b_v1, 0, Vhidb, 64, 0);

    heads_kernel<<<bt, blk, 0, stream>>>(Qhidb, Vhidb, W_q2, b_q2, actions, W_v2, b_v2, Qv, V);
}